// NetworkLayer_86895778332889
// MI455X (gfx1250) — compile-verified
//
#include <hip/hip_runtime.h>
#include <hip/hip_bf16.h>
#include <math.h>

// ---- problem dims (from reference) ----
#define DMODEL 1024
#define DINNER 2048        // 2*DMODEL
#define DSTATE 16
#define DCONV  4
#define DTRANK 64          // DMODEL/16
#define BATCH  2
#define SEQL   2048
#define NTOK   (BATCH*SEQL)   // 4096
#define XZ_LD  (2*DINNER)     // 4096
#define XDBL_LD 96            // DTRANK + 2*DSTATE

typedef float v2f __attribute__((ext_vector_type(2)));
typedef float v8f __attribute__((ext_vector_type(8)));

// ---------------------------------------------------------------------------
// LayerNorm (one block per token) + residual passthrough into d_out tail.
// ---------------------------------------------------------------------------
__global__ void ln_residual_kernel(const float* __restrict__ x,
                                   const float* __restrict__ w,
                                   const float* __restrict__ b,
                                   float* __restrict__ h,
                                   float* __restrict__ resid) {
  const int t = blockIdx.x;
  const float* xr = x + (size_t)t * DMODEL;
  __shared__ float red[256];

  float s = 0.f;
  for (int i = threadIdx.x; i < DMODEL; i += 256) s += xr[i];
  red[threadIdx.x] = s; __syncthreads();
  for (int off = 128; off > 0; off >>= 1) {
    if (threadIdx.x < off) red[threadIdx.x] += red[threadIdx.x + off];
    __syncthreads();
  }
  const float mu = red[0] * (1.f / DMODEL);
  __syncthreads();

  float v = 0.f;
  for (int i = threadIdx.x; i < DMODEL; i += 256) { float d = xr[i] - mu; v += d * d; }
  red[threadIdx.x] = v; __syncthreads();
  for (int off = 128; off > 0; off >>= 1) {
    if (threadIdx.x < off) red[threadIdx.x] += red[threadIdx.x + off];
    __syncthreads();
  }
  const float rstd = rsqrtf(red[0] * (1.f / DMODEL) + 1e-5f);

  for (int i = threadIdx.x; i < DMODEL; i += 256) {
    const float xv = xr[i];
    h[(size_t)t * DMODEL + i] = (xv - mu) * rstd * w[i] + b[i];
    resid[(size_t)t * DMODEL + i] = xv;
  }
}

// ---------------------------------------------------------------------------
// Generic fp32 GEMM via V_WMMA_F32_16X16X4_F32.
//   C[M x N] = A[M x K] * W[N x K]^T   (weights row-major [out, in])
// Template NT = 16-wide subtiles per wave; grid/block chosen so tiling is
// EXACT (no runtime guards -> uniform control flow -> EXEC all-1s at WMMA).
// K is chunked by 16: all (4 + 4*NT) fragment loads of a chunk are issued
// into registers BEFORE the 4*NT WMMAs, so loads drain under decreasing
// loadcnt waits instead of wait-0 before every WMMA.
// A fragment (16x4 f32): lane l<16 holds A[m0+l][k..k+1], lane l+16 holds
// A[m0+l][k+2..k+3] -> v2f per documented VGPR layout. B mirrored.
// ---------------------------------------------------------------------------
template <int NT>
__global__ void gemm_wmma_f32(const float* __restrict__ A,
                              const float* __restrict__ W,
                              float* __restrict__ C,
                              int K, int lda, int ldw, int ldc) {
  const int lane = threadIdx.x;        // 0..31
  const int half = lane >> 4;          // 0 or 1
  const int l    = lane & 15;
  const int m0    = blockIdx.y * 16;
  const int nbase = (blockIdx.x * blockDim.y + threadIdx.y) * (16 * NT);

  v8f acc[NT] = {};
  const float* Arow = A + (size_t)(m0 + l) * lda + 2 * half;
  const float* Wcol[NT];
#pragma unroll
  for (int s = 0; s < NT; ++s)
    Wcol[s] = W + (size_t)(nbase + s * 16 + l) * ldw + 2 * half;

  for (int k = 0; k < K; k += 16) {     // K % 16 == 0 for all call sites
    v2f a4[4];
    v2f b4[NT][4];
    // ---- load phase: (4 + 4*NT) outstanding global_load_b64 ----
#pragma unroll
    for (int j = 0; j < 4; ++j)
      a4[j] = *reinterpret_cast<const v2f*>(Arow + k + 4 * j);
#pragma unroll
    for (int s = 0; s < NT; ++s)
#pragma unroll
      for (int j = 0; j < 4; ++j)
        b4[s][j] = *reinterpret_cast<const v2f*>(Wcol[s] + k + 4 * j);
    // ---- compute phase: 4*NT WMMAs ----
#pragma unroll
    for (int j = 0; j < 4; ++j)
#pragma unroll
      for (int s = 0; s < NT; ++s)
        acc[s] = __builtin_amdgcn_wmma_f32_16x16x4_f32(
            false, a4[j], false, b4[s][j], (short)0, acc[s], false, false);
  }

#pragma unroll
  for (int s = 0; s < NT; ++s) {
    const int col = nbase + s * 16 + l;
#pragma unroll
    for (int r = 0; r < 8; ++r) {
      const int row = m0 + half * 8 + r;   // C/D layout: lanes16-31 -> M=8+r
      C[(size_t)row * ldc + col] = acc[s][r];
    }
  }
}

// ---------------------------------------------------------------------------
// Depthwise causal conv1d (width 4) + bias + SiLU. xin = xz[:, 0:DINNER].
// ---------------------------------------------------------------------------
__global__ void conv_silu_kernel(const float* __restrict__ xz,
                                 const float* __restrict__ cw,
                                 const float* __restrict__ cb,
                                 float* __restrict__ u) {
  const size_t idx = (size_t)blockIdx.x * blockDim.x + threadIdx.x;
  if (idx >= (size_t)NTOK * DINNER) return;
  const int d   = (int)(idx % DINNER);
  const int tok = (int)(idx / DINNER);
  const int lp  = tok % SEQL;

  float acc = cb[d];
#pragma unroll
  for (int j = 0; j < DCONV; ++j) {
    const int ll = lp - (DCONV - 1) + j;
    if (ll >= 0)
      acc += cw[d * DCONV + j] * xz[(size_t)(tok - (DCONV - 1) + j) * XZ_LD + d];
  }
  u[idx] = acc / (1.f + __expf(-acc));   // SiLU
}

// ---------------------------------------------------------------------------
// dt = softplus(dt_raw + bias[d])
// ---------------------------------------------------------------------------
__global__ void dt_softplus_kernel(float* __restrict__ dt,
                                   const float* __restrict__ bias) {
  const size_t idx = (size_t)blockIdx.x * blockDim.x + threadIdx.x;
  if (idx >= (size_t)NTOK * DINNER) return;
  const int d = (int)(idx % DINNER);
  const float v = dt[idx] + bias[d];
  dt[idx] = (v > 20.f) ? v : log1pf(__expf(v));
}

// ---------------------------------------------------------------------------
// Selective scan: one lane per (b, d) channel, 16-state in registers,
// sequential over L. Per-step B/C (32 shared floats) staged through LDS.
// Writes y (incl. +u*D and *silu(z)) — may alias u (same-thread same-index
// read-then-write).
// ---------------------------------------------------------------------------
__global__ void scan_kernel(const float* __restrict__ dtb,
                            const float* __restrict__ ub,
                            const float* __restrict__ xdbl,
                            const float* __restrict__ xz,
                            const float* __restrict__ A_log,
                            const float* __restrict__ Dv,
                            float* __restrict__ yb) {
  const int d = (blockIdx.x % (DINNER / 256)) * 256 + threadIdx.x;
  const int b = blockIdx.x / (DINNER / 256);

  float A[DSTATE];
#pragma unroll
  for (int s = 0; s < DSTATE; ++s) A[s] = -__expf(A_log[d * DSTATE + s]);
  const float Dd = Dv[d];

  float st[DSTATE];
#pragma unroll
  for (int s = 0; s < DSTATE; ++s) st[s] = 0.f;

  __shared__ float sBC[2 * DSTATE];

  for (int t = 0; t < SEQL; ++t) {
    const size_t tok = (size_t)b * SEQL + t;
    if (threadIdx.x < 2 * DSTATE)
      sBC[threadIdx.x] = xdbl[tok * XDBL_LD + DTRANK + threadIdx.x];
    __syncthreads();

    const float dt = dtb[tok * DINNER + d];
    const float uu = ub[tok * DINNER + d];
    const float du = dt * uu;
    float y = 0.f;
#pragma unroll
    for (int s = 0; s < DSTATE; ++s) {
      const float dA = __expf(dt * A[s]);
      st[s] = dA * st[s] + du * sBC[s];
      y += st[s] * sBC[DSTATE + s];
    }
    y += uu * Dd;
    const float z = xz[tok * XZ_LD + DINNER + d];
    y *= z / (1.f + __expf(-z));
    yb[tok * DINNER + d] = y;
    __syncthreads();
  }
}

// ---------------------------------------------------------------------------
extern "C" void kernel_launch(void* const* d_in, const int* in_sizes, int n_in,
                              void* d_out, int out_size, void* d_ws, size_t ws_size,
                              hipStream_t stream) {
  const float* x         = (const float*)d_in[0];
  const float* in_proj_w = (const float*)d_in[1];
  const float* conv_w    = (const float*)d_in[2];
  const float* conv_b    = (const float*)d_in[3];
  const float* x_proj_w  = (const float*)d_in[4];
  const float* dt_proj_w = (const float*)d_in[5];
  const float* dt_proj_b = (const float*)d_in[6];
  const float* A_log     = (const float*)d_in[7];
  const float* Dv        = (const float*)d_in[8];
  const float* out_proj_w= (const float*)d_in[9];
  const float* norm_w    = (const float*)d_in[10];
  const float* norm_b    = (const float*)d_in[11];

  float* out   = (float*)d_out;
  float* resid = out + (size_t)NTOK * DMODEL;

  // workspace layout (floats): xz | u(=y) | x_dbl | h (dt aliases h onward)
  float* ws   = (float*)d_ws;
  float* xz   = ws;
  float* u    = xz   + (size_t)NTOK * XZ_LD;
  float* xdbl = u    + (size_t)NTOK * DINNER;
  float* h    = xdbl + (size_t)NTOK * XDBL_LD;
  float* dt   = h;    // h dead after in_proj GEMM; dt extends past it
  float* y    = u;    // in-place over u (safe: per-(t,d) read-then-write)

  const dim3 blk4(32, 4);   // 4 waves * NT=4 * 16 = 256 cols/block
  const dim3 blk2(32, 2);   // 2 waves * NT=3 * 16 = 96  cols/block

  // 1) LayerNorm + residual copy
  ln_residual_kernel<<<NTOK, 256, 0, stream>>>(x, norm_w, norm_b, h, resid);

  // 2) in_proj: xz[4096 x 4096] = h[4096 x 1024] * in_proj_w[4096 x 1024]^T
  gemm_wmma_f32<4><<<dim3(XZ_LD / 256, NTOK / 16), blk4, 0, stream>>>(
      h, in_proj_w, xz, DMODEL, DMODEL, DMODEL, XZ_LD);

  // 3) depthwise conv + SiLU -> u
  {
    const size_t n = (size_t)NTOK * DINNER;
    conv_silu_kernel<<<(unsigned)((n + 255) / 256), 256, 0, stream>>>(
        xz, conv_w, conv_b, u);
  }

  // 4) x_proj: x_dbl[4096 x 96] = u * x_proj_w[96 x 2048]^T   (exact 96-col block)
  gemm_wmma_f32<3><<<dim3(1, NTOK / 16), blk2, 0, stream>>>(
      u, x_proj_w, xdbl, DINNER, DINNER, DINNER, XDBL_LD);

  // 5) dt_proj: dt[4096 x 2048] = x_dbl[:, :64] * dt_proj_w[2048 x 64]^T
  gemm_wmma_f32<4><<<dim3(DINNER / 256, NTOK / 16), blk4, 0, stream>>>(
      xdbl, dt_proj_w, dt, DTRANK, XDBL_LD, DTRANK, DINNER);

  // 6) + bias, softplus
  {
    const size_t n = (size_t)NTOK * DINNER;
    dt_softplus_kernel<<<(unsigned)((n + 255) / 256), 256, 0, stream>>>(dt, dt_proj_b);
  }

  // 7) selective scan (+ u*D, * silu(z)) -> y
  scan_kernel<<<BATCH * (DINNER / 256), 256, 0, stream>>>(
      dt, u, xdbl, xz, A_log, Dv, y);

  // 8) out_proj: out[4096 x 1024] = y * out_proj_w[1024 x 2048]^T
  gemm_wmma_f32<4><<<dim3(DMODEL / 256, NTOK / 16), blk4, 0, stream>>>(
      y, out_proj_w, out, DINNER, DINNER, DINNER, DMODEL);
}